// TaylorAugmented_919123002004
// MI455X (gfx1250) — compile-verified
//
#include <hip/hip_runtime.h>
#include <math.h>

typedef __attribute__((ext_vector_type(2))) float v2f;
typedef __attribute__((ext_vector_type(8))) float v8f;

// ---------------------------------------------------------------------------
// GEMV via V_WMMA_F32_16X16X4_F32:  out[n] += sum_{k in chunk} x[k] * W[k*N+n]
// W row-major (K x N). Each wave owns 4 tiles of 16 output columns (64 cols)
// with 4 independent accumulators -> 8 independent global B-loads per WMMA
// quad (max memory-level parallelism; this kernel is pure HBM streaming).
// A-matrix rows are all the same x-chunk (M dim broadcast-wasted; compute is
// not the bottleneck at 23.3 TB/s).
// A layout (ISA 7.12.2, 32-bit A 16x4): lanes0-15 VGPR0=K0,VGPR1=K1;
//                                       lanes16-31 VGPR0=K2,VGPR1=K3.
// B layout (4x16, rows striped across lanes): VGPR0 lanes0-15=rowK0,
//   lanes16-31=rowK2; VGPR1 = rows K1/K3.
// ---------------------------------------------------------------------------
__global__ void __launch_bounds__(256)
gemv_wmma_kernel(const float* __restrict__ W, const float* __restrict__ x,
                 float* __restrict__ out, int N, int kchunk) {
  __shared__ float xs[1024];
  const int kbase = blockIdx.y * kchunk;
  for (int i = threadIdx.x; i < kchunk; i += 256) xs[i] = x[kbase + i];
  __syncthreads();

  const int wave = threadIdx.x >> 5;
  const int lane = threadIdx.x & 31;
  const int n0   = (blockIdx.x * 8 + wave) * 64 + (lane & 15);
  const int krow = (lane >> 4) << 1;  // 0 for lanes 0-15, 2 for lanes 16-31

  const float* Wp = W + (size_t)(kbase + krow) * N + n0;
  v8f acc0 = {}, acc1 = {}, acc2 = {}, acc3 = {};

  for (int k = 0; k < kchunk; k += 4) {
    if (((k & 31) == 0) && (k + 32 < kchunk))
      __builtin_prefetch(Wp + (size_t)(k + 32) * N, 0, 1);  // global_prefetch_b8
    v2f a; a.x = xs[k + krow]; a.y = xs[k + krow + 1];
    const float* r0 = Wp + (size_t)k * N;
    const float* r1 = r0 + N;
    v2f b0; b0.x = r0[0];  b0.y = r1[0];
    v2f b1; b1.x = r0[16]; b1.y = r1[16];
    v2f b2; b2.x = r0[32]; b2.y = r1[32];
    v2f b3; b3.x = r0[48]; b3.y = r1[48];
    // 8 args: (neg_a, A, neg_b, B, c_mod, C, reuse_a, reuse_b)
    acc0 = __builtin_amdgcn_wmma_f32_16x16x4_f32(false, a, false, b0, (short)0, acc0, false, false);
    acc1 = __builtin_amdgcn_wmma_f32_16x16x4_f32(false, a, false, b1, (short)0, acc1, false, false);
    acc2 = __builtin_amdgcn_wmma_f32_16x16x4_f32(false, a, false, b2, (short)0, acc2, false, false);
    acc3 = __builtin_amdgcn_wmma_f32_16x16x4_f32(false, a, false, b3, (short)0, acc3, false, false);
  }
  // All D rows identical (A rows identical); take M=0 row: VGPR0, lanes 0-15.
  if (lane < 16) {
    atomicAdd(&out[n0],      acc0[0]);
    atomicAdd(&out[n0 + 16], acc1[0]);
    atomicAdd(&out[n0 + 32], acc2[0]);
    atomicAdd(&out[n0 + 48], acc3[0]);
  }
}

// ---------------------------- elementwise helpers ---------------------------
__global__ void zero_kernel(float* __restrict__ p, int n) {
  int i = blockIdx.x * blockDim.x + threadIdx.x;
  if (i < n) p[i] = 0.0f;
}

__global__ void concat2_kernel(const float* __restrict__ a, const float* __restrict__ b,
                               float* __restrict__ o, int na, int nb) {
  int i = blockIdx.x * blockDim.x + threadIdx.x;
  if (i < na) o[i] = a[i];
  else if (i < na + nb) o[i] = b[i - na];
}

__global__ void tanh_u_kernel(const float* __restrict__ a0, const float* __restrict__ b1,
                              float* __restrict__ s0, float* __restrict__ u, int n) {
  int i = blockIdx.x * blockDim.x + threadIdx.x;
  if (i < n) { float s = tanhf(a0[i] + b1[i]); s0[i] = s; u[i] = 1.0f - s * s; }
}

__global__ void add_bias_kernel(const float* __restrict__ x, const float* __restrict__ b,
                                float* __restrict__ o, int n) {
  int i = blockIdx.x * blockDim.x + threadIdx.x;
  if (i < n) o[i] = x[i] + b[i];
}

__global__ void g1_kernel(const float* __restrict__ u, const float* __restrict__ c1,
                          float* __restrict__ g1, int n) {
  int i = blockIdx.x * blockDim.x + threadIdx.x;
  if (i < n) g1[i] = u[i] * c1[i];
}

// g2 = u*c2 - s0*g1*c1   (order-2 tanh jet coefficient)
__global__ void g2_kernel(const float* __restrict__ u, const float* __restrict__ s0,
                          const float* __restrict__ g1, const float* __restrict__ c1,
                          const float* __restrict__ c2, float* __restrict__ g2, int n) {
  int i = blockIdx.x * blockDim.x + threadIdx.x;
  if (i < n) g2[i] = u[i] * c2[i] - s0[i] * g1[i] * c1[i];
}

__global__ void msq_reduce_kernel(const float* __restrict__ z, float* __restrict__ out, int n) {
  __shared__ float red[256];
  float s = 0.0f;
  for (int i = threadIdx.x; i < n; i += 256) { float v = z[i]; s += v * v; }
  red[threadIdx.x] = s;
  __syncthreads();
  for (int off = 128; off > 0; off >>= 1) {
    if ((int)threadIdx.x < off) red[threadIdx.x] += red[threadIdx.x + off];
    __syncthreads();
  }
  if (threadIdx.x == 0) out[0] = red[0] / (float)n;
}

// ---------------------------------------------------------------------------
extern "C" void kernel_launch(void* const* d_in, const int* in_sizes, int n_in,
                              void* d_out, int out_size, void* d_ws, size_t ws_size,
                              hipStream_t stream) {
  // setup_inputs order: h, r, t, c, W1, b1, W2, b2
  const float* h  = (const float*)d_in[0];
  const float* c  = (const float*)d_in[3];
  const float* W1 = (const float*)d_in[4];  // (8192, 8192) row-major
  const float* b1 = (const float*)d_in[5];  // (8192,)
  const float* W2 = (const float*)d_in[6];  // (8192, 4096) row-major
  const float* b2 = (const float*)d_in[7];  // (4096,)
  float* out = (float*)d_out;               // [0..4095]=dydt, [4096]=mean(drdt^2)

  float* ws    = (float*)d_ws;
  float* a0acc = ws;             // 8192  (zeroed)
  float* pacc  = a0acc + 8192;   // 4096  (zeroed)
  float* c1acc = pacc  + 4096;   // 8192  (zeroed)
  float* r1acc = c1acc + 8192;   // 4096  (zeroed)
  float* c2acc = r1acc + 4096;   // 8192  (zeroed)
  float* z2acc = c2acc + 8192;   // 4096  (zeroed)  -- acc region = 36864 floats
  float* xcat  = z2acc + 4096;   // 8192
  float* s0    = xcat  + 8192;   // 8192
  float* u     = s0    + 8192;   // 8192
  float* g1    = u     + 8192;   // 8192
  float* g2    = g1    + 8192;   // 8192

  const dim3 blk(256);

  // zero the atomic accumulators (every call; graph-replay safe)
  zero_kernel<<<dim3((36864 + 255) / 256), blk, 0, stream>>>(ws, 36864);

  // xcat = [h; c]
  concat2_kernel<<<dim3(8192 / 256), blk, 0, stream>>>(h, c, xcat, 4096, 4096);

  // a0 = xcat @ W1        (K=8192, N=8192)
  gemv_wmma_kernel<<<dim3(8192 / 512, 32), blk, 0, stream>>>(W1, xcat, a0acc, 8192, 256);
  // s0 = tanh(a0 + b1); u = 1 - s0^2
  tanh_u_kernel<<<dim3(8192 / 256), blk, 0, stream>>>(a0acc, b1, s0, u, 8192);

  // p = s0 @ W2 + b2      (K=8192, N=4096) -> dydt
  gemv_wmma_kernel<<<dim3(4096 / 512, 32), blk, 0, stream>>>(W2, s0, pacc, 4096, 256);
  add_bias_kernel<<<dim3(4096 / 256), blk, 0, stream>>>(pacc, b2, out, 4096);

  // c1 = p @ W1top        (K=4096, N=8192; top half of W1 = rows 0..4095)
  gemv_wmma_kernel<<<dim3(8192 / 512, 32), blk, 0, stream>>>(W1, out, c1acc, 8192, 128);
  g1_kernel<<<dim3(8192 / 256), blk, 0, stream>>>(u, c1acc, g1, 8192);

  // r1 = g1 @ W2          (K=8192, N=4096)
  gemv_wmma_kernel<<<dim3(4096 / 512, 32), blk, 0, stream>>>(W2, g1, r1acc, 4096, 256);

  // c2 = r1 @ W1top       (K=4096, N=8192)
  gemv_wmma_kernel<<<dim3(8192 / 512, 32), blk, 0, stream>>>(W1, r1acc, c2acc, 8192, 128);
  g2_kernel<<<dim3(8192 / 256), blk, 0, stream>>>(u, s0, g1, c1acc, c2acc, g2, 8192);

  // z2 = g2 @ W2          (K=8192, N=4096)
  gemv_wmma_kernel<<<dim3(4096 / 512, 32), blk, 0, stream>>>(W2, g2, z2acc, 4096, 256);

  // mean(z2^2) -> out[4096]
  msq_reduce_kernel<<<dim3(1), blk, 0, stream>>>(z2acc, out + 4096, 4096);
}